// CustomQuaternionLoss_35115652612310
// MI455X (gfx1250) — compile-verified
//
#include <hip/hip_runtime.h>
#include <math.h>

#define DT_STEP 0.01f
#define EPS_V   1e-6f

typedef float v2f __attribute__((ext_vector_type(2)));
typedef float v8f __attribute__((ext_vector_type(8)));

struct Quat { float w, x, y, z; };  // reference order [w, x, y, z]

__device__ __forceinline__ Quat qmul(const Quat q, const Quat r) {
    Quat o;
    o.w = q.w * r.w - q.x * r.x - q.y * r.y - q.z * r.z;
    o.x = q.w * r.x + q.x * r.w + q.y * r.z - q.z * r.y;
    o.y = q.w * r.y - q.x * r.z + q.y * r.w + q.z * r.x;
    o.z = q.w * r.z + q.x * r.y - q.y * r.x + q.z * r.w;
    return o;
}

__device__ __forceinline__ float huber1(float x) {
    float a = fabsf(x);
    return (a <= 1.0f) ? (0.5f * x * x) : (a - 0.5f);
}

// Exact 32-lane sum using the CDNA5 matrix pipe.
// A-layout (f32 16x4): lanes 0-15 -> A[lane, 0] (vgpr0), lanes 16-31 -> A[lane-16, 2] (vgpr0);
// with a[1]=0 the 32 partials fill A, rest is zero. B = ones(4x16) =>
// D[m,n] = p_m + p_{m+16}. Summing the 8 C/D vgprs of a lane gives the sum of
// 8 such rows (rows 0-7 for lanes 0-15, rows 8-15 for lanes 16-31); one
// shfl_xor(16) combines the two halves -> full wave sum in every lane.
__device__ __forceinline__ float wave_reduce_wmma(float p) {
    v2f a; a[0] = p;    a[1] = 0.0f;
    v2f b; b[0] = 1.0f; b[1] = 1.0f;
    v8f c = {};
    c = __builtin_amdgcn_wmma_f32_16x16x4_f32(
        /*neg_a=*/false, a, /*neg_b=*/false, b,
        /*c_mod=*/(short)0, c, /*reuse_a=*/false, /*reuse_b=*/false);
    float s = c[0] + c[1] + c[2] + c[3] + c[4] + c[5] + c[6] + c[7];
    s += __shfl_xor(s, 16, 32);
    return s;
}

__global__ void ql_init_kernel(float* acc) {
    if (blockIdx.x == 0 && threadIdx.x == 0) acc[0] = 0.0f;
}

__global__ __launch_bounds__(256) void ql_main_kernel(
    const float* __restrict__ true_q,   // (B,4)
    const float* __restrict__ biases,   // (B,3)
    const float* __restrict__ batch_X,  // (B,T,6)
    const float* __restrict__ quats,    // (N,4)
    const int*   __restrict__ indices,  // (B,)
    const int*   __restrict__ seq_len,  // scalar (device)
    float* __restrict__ acc,
    int B)
{
    const int T   = seq_len[0];                 // uniform scalar load
    const int tid = blockIdx.x * blockDim.x + threadIdx.x;

    float partial = 0.0f;
    if (tid < B) {
        // ang = batch_X[b, T-1, 3:6]
        const float* xrow = batch_X + (size_t)tid * (size_t)(T * 6)
                                    + (size_t)(T - 1) * 6 + 3;
        const float ax = xrow[0], ay = xrow[1], az = xrow[2];

        const float* bb = biases + (size_t)tid * 3;
        const float wx = ax - bb[0];
        const float wy = ay - bb[1];
        const float wz = az - bb[2];

        const float wn = sqrtf(wx * wx + wy * wy + wz * wz);
        const float ht = 0.5f * wn * DT_STEP;
        float sh, ch;
        sincosf(ht, &sh, &ch);
        const float iwn = 1.0f / wn;            // matches reference (no clamp here)
        Quat rot = { ch, sh * wx * iwn, sh * wy * iwn, sh * wz * iwn };
        {
            const float rn = 1.0f / sqrtf(rot.w*rot.w + rot.x*rot.x +
                                          rot.y*rot.y + rot.z*rot.z);
            rot.w *= rn; rot.x *= rn; rot.y *= rn; rot.z *= rn;
        }

        // q0 gather (table is L2-resident) + normalize
        int ii = indices[tid] - (T - 1);
        ii = (ii < 0) ? 0 : ii;
        const float4 qv = reinterpret_cast<const float4*>(quats)[ii];
        Quat q0 = { qv.x, qv.y, qv.z, qv.w };
        {
            const float qn = 1.0f / sqrtf(q0.w*q0.w + q0.x*q0.x +
                                          q0.y*q0.y + q0.z*q0.z);
            q0.w *= qn; q0.x *= qn; q0.y *= qn; q0.z *= qn;
        }

        const Quat comp = qmul(q0, rot);
        const Quat ccnj = { comp.w, -comp.x, -comp.y, -comp.z };

        const float4 tv = reinterpret_cast<const float4*>(true_q)[tid];
        const Quat tq = { tv.x, tv.y, tv.z, tv.w };

        Quat diff = qmul(ccnj, tq);
        {
            const float dn = 1.0f / sqrtf(diff.w*diff.w + diff.x*diff.x +
                                          diff.y*diff.y + diff.z*diff.z);
            diff.w *= dn; diff.x *= dn; diff.y *= dn; diff.z *= dn;
        }

        float vn = sqrtf(diff.x*diff.x + diff.y*diff.y + diff.z*diff.z);
        vn = (vn > EPS_V) ? vn : EPS_V;
        const float angle = 2.0f * atan2f(vn, diff.w);
        const float sc = angle / vn;

        partial = huber1(sc * diff.x) + huber1(sc * diff.y) + huber1(sc * diff.z);
    }
    // guard re-converges here: EXEC is all-1s at the WMMA (ISA requirement)

    const float wsum = wave_reduce_wmma(partial);

    __shared__ float lsum[8];                   // 256 threads = 8 wave32
    const int lane = threadIdx.x & 31;
    const int wid  = threadIdx.x >> 5;
    if (lane == 0) lsum[wid] = wsum;
    __syncthreads();
    if (threadIdx.x == 0) {
        float t = lsum[0] + lsum[1] + lsum[2] + lsum[3]
                + lsum[4] + lsum[5] + lsum[6] + lsum[7];
        atomicAdd(acc, t);                      // one global_atomic_add_f32 / block
    }
}

__global__ void ql_final_kernel(const float* __restrict__ acc,
                                float* __restrict__ out,
                                float inv_count) {
    if (blockIdx.x == 0 && threadIdx.x == 0) out[0] = acc[0] * inv_count;
}

extern "C" void kernel_launch(void* const* d_in, const int* in_sizes, int n_in,
                              void* d_out, int out_size, void* d_ws, size_t ws_size,
                              hipStream_t stream) {
    const float* true_q  = (const float*)d_in[0];
    const float* biases  = (const float*)d_in[1];
    const float* batch_X = (const float*)d_in[2];
    const float* quats   = (const float*)d_in[3];
    const int*   indices = (const int*)  d_in[4];
    const int*   seq_len = (const int*)  d_in[5];

    const int B = in_sizes[0] / 4;              // true_quaternions is (B,4)
    float* acc = (float*)d_ws;
    float* out = (float*)d_out;

    ql_init_kernel<<<1, 64, 0, stream>>>(acc);

    const int blocks = (B + 255) / 256;
    ql_main_kernel<<<blocks, 256, 0, stream>>>(true_q, biases, batch_X, quats,
                                               indices, seq_len, acc, B);

    ql_final_kernel<<<1, 64, 0, stream>>>(acc, out, 1.0f / (3.0f * (float)B));
}